// VPNNetwork_53833120088716
// MI455X (gfx1250) — compile-verified
//
#include <hip/hip_runtime.h>
#include <hip/hip_bf16.h>

typedef float v2f __attribute__((ext_vector_type(2)));
typedef float v8f __attribute__((ext_vector_type(8)));

#define BK 128              // batch rows per block (4 waves x 32 rows)
#define STR 49              // padded LDS row stride (gcd(49,64)=1 -> conflict-free)

__global__ __launch_bounds__(BK) void vpn_fused_kernel(
    const float* __restrict__ obs,      // (B,48)
    const float* __restrict__ W_phi,    // (48,48)
    const float* __restrict__ b_phi,    // (48,)
    const float* __restrict__ W_logit,  // (4,36)
    const float* __restrict__ b_logit,  // (4,)
    float* __restrict__ out,            // [B*4 logit | B V_]
    int B)
{
    __shared__ float s_obs[BK * STR];   // obs tile, row-major padded
    __shared__ float s_phi[BK * STR];   // phi result -> later reused as padded-V scratch
    __shared__ float s_w[48 * STR];     // W_phi, row-major padded (row n = output feature)

    const int tid = threadIdx.x;
    const int b0  = blockIdx.x * BK;

    // ---- stage W_phi into LDS (padded) ----
    for (int idx = tid; idx < 48 * 48; idx += BK)
        s_w[(idx / 48) * STR + (idx % 48)] = W_phi[idx];

    // ---- stage obs tile (coalesced float4 loads) ----
    {
        const float4* obs4 = reinterpret_cast<const float4*>(obs + (size_t)b0 * 48);
        for (int c = tid; c < BK * 12; c += BK) {   // 12 float4 per 48-float row
            int r = c / 12, q = c % 12;
            float4 val;
            if (b0 + r < B) val = obs4[(size_t)r * 12 + q];
            else            val = make_float4(0.f, 0.f, 0.f, 0.f);
            int base = r * STR + q * 4;
            s_obs[base + 0] = val.x; s_obs[base + 1] = val.y;
            s_obs[base + 2] = val.z; s_obs[base + 3] = val.w;
        }
    }
    __syncthreads();

    // ---- stage 1: phi = obs @ W_phi.T + b_phi via V_WMMA_F32_16X16X4_F32 ----
    const int ln = tid & 31;
    const int hi = ln >> 4;       // half-wave select
    const int lm = ln & 15;
    const int m0 = (tid >> 5) * 32;   // this wave's 32-row slice
    const int m1 = m0 + 16;

    v8f acc[2][3];
    #pragma unroll
    for (int mt = 0; mt < 2; ++mt)
        #pragma unroll
        for (int t = 0; t < 3; ++t)
            #pragma unroll
            for (int r = 0; r < 8; ++r) acc[mt][t][r] = 0.0f;

    #pragma unroll
    for (int kk = 0; kk < 12; ++kk) {
        const int k0 = kk * 4 + 2 * hi;
        v2f a0, a1;
        a0.x = s_obs[(m0 + lm) * STR + k0];
        a0.y = s_obs[(m0 + lm) * STR + k0 + 1];
        a1.x = s_obs[(m1 + lm) * STR + k0];
        a1.y = s_obs[(m1 + lm) * STR + k0 + 1];
        #pragma unroll
        for (int t = 0; t < 3; ++t) {
            v2f bb;   // B[k][n] = W_phi[n][k]
            bb.x = s_w[(t * 16 + lm) * STR + k0];
            bb.y = s_w[(t * 16 + lm) * STR + k0 + 1];
            acc[0][t] = __builtin_amdgcn_wmma_f32_16x16x4_f32(
                false, a0, false, bb, (short)0, acc[0][t], false, false);
            acc[1][t] = __builtin_amdgcn_wmma_f32_16x16x4_f32(
                false, a1, false, bb, (short)0, acc[1][t], false, false);
        }
    }

    // bias + transpose through LDS (row = batch row, col = feature)
    #pragma unroll
    for (int t = 0; t < 3; ++t) {
        const float bias = b_phi[t * 16 + lm];
        #pragma unroll
        for (int r = 0; r < 8; ++r) {
            s_phi[(m0 + r + 8 * hi) * STR + (t * 16 + lm)] = acc[0][t][r] + bias;
            s_phi[(m1 + r + 8 * hi) * STR + (t * 16 + lm)] = acc[1][t][r] + bias;
        }
    }
    __syncthreads();

    // ---- stage 2: per-lane Bellman iteration (one batch row per thread) ----
    const int pb = tid * STR;   // this lane's phi row in LDS
    const int ob = tid * STR;   // this lane's obs row in LDS

    float rin[16], rou[16], p[16];
    #pragma unroll
    for (int c = 0; c < 16; ++c) {
        rin[c] = s_phi[pb + 3 * c + 0];
        rou[c] = s_phi[pb + 3 * c + 1];
        p[c]   = s_phi[pb + 3 * c + 2];
    }

    constexpr int DI[4] = {-1, 1, 0, 0};
    constexpr int DJ[4] = {0, 0, -1, 1};

    // iteration-invariant edge constants: e = rin[n] - (rin[n]!=0)*rout[c]
    float e[16][4];
    #pragma unroll
    for (int c = 0; c < 16; ++c)
        #pragma unroll
        for (int d = 0; d < 4; ++d) e[c][d] = 0.0f;
    #pragma unroll
    for (int i = 0; i < 4; ++i)
        #pragma unroll
        for (int j = 0; j < 4; ++j)
            #pragma unroll
            for (int d = 0; d < 4; ++d) {
                const int ni = i + DI[d], nj = j + DJ[d];
                if (ni >= 0 && ni < 4 && nj >= 0 && nj < 4) {
                    const int c = i * 4 + j, n = ni * 4 + nj;
                    e[c][d] = rin[n] - (rin[n] != 0.0f ? rou[c] : 0.0f);
                }
            }

    float v[16];
    #pragma unroll
    for (int c = 0; c < 16; ++c) v[c] = 0.0f;

    for (int it = 0; it < 20; ++it) {   // all offsets read start-of-step v
        float vn[16];
        #pragma unroll
        for (int i = 0; i < 4; ++i)
            #pragma unroll
            for (int j = 0; j < 4; ++j) {
                const int c = i * 4 + j;
                float m = v[c];
                #pragma unroll
                for (int d = 0; d < 4; ++d) {
                    const int ni = i + DI[d], nj = j + DJ[d];
                    if (ni >= 0 && ni < 4 && nj >= 0 && nj < 4)
                        m = fmaxf(m, fmaf(p[c], v[ni * 4 + nj], e[c][d]));
                }
                vn[c] = m;
            }
        #pragma unroll
        for (int c = 0; c < 16; ++c) v[c] = vn[c];
    }

    // ---- stage 3: agent position, padded-V to LDS, gathers, logits ----
    // reuse this lane's phi row as zero-padded 6x6 V (only own row touched;
    // same-wave DS ops are in-order, so no barrier needed)
    #pragma unroll
    for (int idx = 0; idx < 36; ++idx) {
        const int r = idx / 6, c = idx % 6;
        const float val = (r >= 1 && r <= 4 && c >= 1 && c <= 4)
                              ? v[(r - 1) * 4 + (c - 1)] : 0.0f;
        s_phi[pb + idx] = val;
    }

    int  first = 0;
    bool has   = false;
    #pragma unroll
    for (int c = 15; c >= 0; --c) {     // descending -> keeps smallest index
        const float ch1 = s_obs[ob + 3 * c + 1];
        if (ch1 != 0.0f) { first = c; has = true; }
    }
    const int i_ = has ? (first >> 2) : 1;
    const int j_ = has ? (first & 3) : 1;

    float sv[9];
    #pragma unroll
    for (int dr = 0; dr < 3; ++dr)
        #pragma unroll
        for (int dc = 0; dc < 3; ++dc)
            sv[dr * 3 + dc] = s_phi[pb + (i_ + dr) * 6 + (j_ + dc)];
    const float Vstar = sv[4];

    float ss[27];
    #pragma unroll
    for (int dr = 0; dr < 3; ++dr)
        #pragma unroll
        for (int dc = 0; dc < 3; ++dc) {
            const int r = i_ - 1 + dr, c = j_ - 1 + dc;
            const bool inb = (r >= 0 && r < 4 && c >= 0 && c < 4);
            const int cell = inb ? (r * 4 + c) : 0;
            #pragma unroll
            for (int ch = 0; ch < 3; ++ch)
                ss[dr * 9 + dc * 3 + ch] = inb ? s_obs[ob + 3 * cell + ch] : 0.0f;
        }

    float lg[4];
    #pragma unroll
    for (int o = 0; o < 4; ++o) {
        float a = b_logit[o];              // uniform addresses -> s_load
        #pragma unroll
        for (int f = 0; f < 27; ++f) a = fmaf(W_logit[o * 36 + f], ss[f], a);
        #pragma unroll
        for (int f = 0; f < 9; ++f)  a = fmaf(W_logit[o * 36 + 27 + f], sv[f], a);
        lg[o] = a;
    }

    const int gb = b0 + tid;
    if (gb < B) {
        float4 lgv = make_float4(lg[0], lg[1], lg[2], lg[3]);
        *reinterpret_cast<float4*>(out + (size_t)gb * 4) = lgv;
        out[(size_t)4 * B + gb] = Vstar;
    }
}

extern "C" void kernel_launch(void* const* d_in, const int* in_sizes, int n_in,
                              void* d_out, int out_size, void* d_ws, size_t ws_size,
                              hipStream_t stream) {
    (void)n_in; (void)out_size; (void)d_ws; (void)ws_size;
    const float* obs     = (const float*)d_in[0];
    const float* W_phi   = (const float*)d_in[1];
    const float* b_phi   = (const float*)d_in[2];
    const float* W_logit = (const float*)d_in[3];
    const float* b_logit = (const float*)d_in[4];
    float* out = (float*)d_out;
    const int B = in_sizes[0] / 48;
    const int grid = (B + BK - 1) / BK;
    vpn_fused_kernel<<<grid, BK, 0, stream>>>(obs, W_phi, b_phi, W_logit, b_logit, out, B);
}